// SpatialMamba_22943715295255
// MI455X (gfx1250) — compile-verified
//
#include <hip/hip_runtime.h>
#include <math.h>

#define D_MODEL 32
#define D_STATE 64
#define D_CONV  4
#define D_INNER 64
#define BATCH   4
#define HH      48
#define WW      48
#define SEQL    48
#define NSEQ    (BATCH * WW)   /* 192, same count for vertical (BATCH*HH) */
#define BLOCK   128

typedef float v2f __attribute__((ext_vector_type(2)));
typedef float v8f __attribute__((ext_vector_type(8)));

static __device__ __forceinline__ v8f wmma4(v2f a, v2f b, v8f c) {
  // D = A(16x4,f32) * B(4x16,f32) + C ; exact f32, CDNA5 V_WMMA_F32_16X16X4_F32
  return __builtin_amdgcn_wmma_f32_16x16x4_f32(false, a, false, b, (short)0, c,
                                               false, false);
}

// One 16x16 f32 tile of  C = A * W^T.
// A: row-major [M x K] (LDS), lda floats per row.
// W: row-major [N x K] (global), ldw floats per row (so B[k][n] = W[n][k]).
static __device__ __forceinline__ v8f gemm_tile(const float* __restrict__ A, int lda, int M0,
                                                const float* __restrict__ W, int ldw, int N0,
                                                int K, int lane) {
  const int m  = lane & 15;          // A: M index / B: N index
  const int kh = (lane >> 4) << 1;   // lanes 16-31 hold K rows v+2
  v8f c = {};
  for (int k0 = 0; k0 < K; k0 += 4) {
    v2f a, b;
    a.x = A[(M0 + m) * lda + k0 + kh];
    a.y = A[(M0 + m) * lda + k0 + kh + 1];
    b.x = W[(N0 + m) * ldw + k0 + kh];
    b.y = W[(N0 + m) * ldw + k0 + kh + 1];
    c = wmma4(a, b, c);
  }
  return c;
}

__global__ __launch_bounds__(BLOCK) void spatial_mamba_dir(
    const float* __restrict__ x,      const float* __restrict__ W_in,
    const float* __restrict__ conv_w, const float* __restrict__ conv_b,
    const float* __restrict__ W_x,    const float* __restrict__ W_dt,
    const float* __restrict__ b_dt,   const float* __restrict__ A_log,
    const float* __restrict__ Dvec,   const float* __restrict__ W_out,
    float* __restrict__ out)
{
  // 48 * (64 + 64 + 192) floats = 61440 B static LDS
  __shared__ float smem[SEQL * 320];
  float* s_xs = smem;               // [48][64]  x_ssm (pre-conv, then conv+silu)
  float* s_z  = smem + SEQL * 64;   // [48][64]  z ; later y_final
  float* s_sp = smem + SEQL * 128;  // [48][192] d_in|Bp|Cp ; d_in -> delta
  float* s_x  = s_sp;               // [48][32]  input staging (aliased, dead before sp)

  const int tid  = threadIdx.x;
  const int lane = tid & 31;
  // wave index is wave-uniform: pin to SGPR so tile loops / store splits are scalar
  const int wave = __builtin_amdgcn_readfirstlane(tid >> 5);
  const int dir  = blockIdx.x / NSEQ;   // 0:h-fwd 1:h-bwd 2:v-fwd 3:v-bwd
  const int seq  = blockIdx.x % NSEQ;
  const int b    = seq / 48;
  const int p    = seq % 48;            // w for horizontal scans, h for vertical
  const bool rev  = (dir & 1);
  const bool vert = (dir >= 2);

  // ---- load input sequence (reversed for backward dirs so scan runs forward) ----
  for (int i = tid; i < SEQL * D_MODEL; i += BLOCK) {
    int l = i >> 5, c = i & 31;
    int gl = rev ? (SEQL - 1 - l) : l;
    int gidx = vert ? (((b * HH + p) * WW + gl) * D_MODEL + c)
                    : (((b * HH + gl) * WW + p) * D_MODEL + c);
    s_x[l * D_MODEL + c] = x[gidx];
  }
  __syncthreads();

  // ---- GEMM1: [48,32] @ W_in^T[32,128] -> x_ssm (cols 0..63) | z (cols 64..127)
  for (int t = wave; t < 24; t += 4) {
    int mt = t >> 3, nt = t & 7;               // scalar (wave-uniform)
    v8f c = gemm_tile(s_x, D_MODEL, mt * 16, W_in, D_MODEL, nt * 16, D_MODEL, lane);
    float* dst = (nt < 4) ? s_xs : s_z;        // scalar select, no lane divergence
    int colbase = (nt < 4) ? nt * 16 : (nt - 4) * 16;
    int col = colbase + (lane & 15);
#pragma unroll
    for (int i = 0; i < 8; ++i) {
      int row = mt * 16 + i + ((lane >> 4) << 3);
      dst[row * 64 + col] = c[i];
    }
  }
  __syncthreads();

  // ---- causal depthwise conv (k=4) + SiLU, in place via registers ----
  {
    float tmp[24];  // 48*64 / 128 threads = 24 elements each
#pragma unroll
    for (int n = 0; n < 24; ++n) {
      int i = tid + n * BLOCK;
      int l = i >> 6, d = i & 63;
      float acc = conv_b[d];
#pragma unroll
      for (int k = 0; k < D_CONV; ++k) {
        int ls = l + k - (D_CONV - 1);
        if (ls >= 0) acc = fmaf(s_xs[ls * 64 + d], conv_w[d * D_CONV + k], acc);
      }
      tmp[n] = acc;
    }
    __syncthreads();
#pragma unroll
    for (int n = 0; n < 24; ++n) {
      float v = tmp[n];
      s_xs[tid + n * BLOCK] = v / (1.f + __expf(-v));   // silu (native exp)
    }
  }
  __syncthreads();

  // ---- GEMM2: [48,64] @ W_x^T[64,192] -> d_in | Bp | Cp ----
  for (int t = wave; t < 36; t += 4) {
    int mt = t / 12, nt = t % 12;
    v8f c = gemm_tile(s_xs, 64, mt * 16, W_x, 64, nt * 16, 64, lane);
    int col = nt * 16 + (lane & 15);
#pragma unroll
    for (int i = 0; i < 8; ++i) {
      int row = mt * 16 + i + ((lane >> 4) << 3);
      s_sp[row * 192 + col] = c[i];
    }
  }
  __syncthreads();

  // ---- GEMM3: delta = softplus(d_in @ W_dt^T + b_dt), overwrites d_in region ----
  {
    v8f dacc[3]; int dmt[3], dnt[3]; int nd = 0;
    for (int t = wave; t < 12; t += 4) {
      int mt = t >> 2, nt = t & 3;
      dacc[nd] = gemm_tile(s_sp, 192, mt * 16, W_dt, 64, nt * 16, 64, lane);
      dmt[nd] = mt; dnt[nd] = nt; ++nd;
    }
    __syncthreads();   // all d_in reads complete before overwrite
    for (int q = 0; q < nd; ++q) {
      int col = dnt[q] * 16 + (lane & 15);
      float bd = b_dt[col];
#pragma unroll
      for (int i = 0; i < 8; ++i) {
        int row = dmt[q] * 16 + i + ((lane >> 4) << 3);
        float v = dacc[q][i] + bd;
        s_sp[row * 192 + col] = (v > 20.f) ? v : log1pf(__expf(v));
      }
    }
  }
  __syncthreads();

  // ---- selective scan: wave owns 16 d_inner channels; lane pairs split d_state ----
  {
    const int di   = wave * 16 + (lane & 15);
    const int half = lane >> 4;        // 0: states 0..31, 1: states 32..63
    const int dsb  = half * 32;
    float An[32], h[32];
#pragma unroll
    for (int j = 0; j < 32; ++j) {
      An[j] = -__expf(A_log[di * D_STATE + dsb + j]);
      h[j]  = 0.f;
    }
    const float Dv = Dvec[di];
    for (int l = 0; l < SEQL; ++l) {
      const float* row = s_sp + l * 192;
      float delta = row[di];                 // softplussed
      float Bb    = delta * row[64 + di];    // B_bar = delta * Bp
      float y     = 0.f;
#pragma unroll
      for (int j = 0; j < 32; ++j) {
        float a = __expf(delta * An[j]);     // A_bar, arg <= 0 -> native exp safe
        h[j] = fmaf(a, h[j], Bb);
        y    = fmaf(h[j], row[128 + dsb + j], y);   // * Cp
      }
      y += __shfl_xor(y, 16, 32);            // reduce the two state halves in-wave
      if (half == 0) {
        float xc = s_xs[l * 64 + di];
        float zv = s_z [l * 64 + di];
        float g  = zv / (1.f + __expf(-zv)); // silu gate
        s_z[l * 64 + di] = (y + Dv * xc) * g;  // y_final overwrites dead z
      }
    }
  }
  __syncthreads();

  // ---- GEMM4: [48,64] @ W_out^T[64,32], atomic-accumulate 4 directions ----
  for (int t = wave; t < 6; t += 4) {
    int mt = t >> 1, nt = t & 1;
    v8f c = gemm_tile(s_z, 64, mt * 16, W_out, 64, nt * 16, 64, lane);
    int col = nt * 16 + (lane & 15);
#pragma unroll
    for (int i = 0; i < 8; ++i) {
      int l  = mt * 16 + i + ((lane >> 4) << 3);
      int gl = rev ? (SEQL - 1 - l) : l;
      int gidx = vert ? (((b * HH + p) * WW + gl) * D_MODEL + col)
                      : (((b * HH + gl) * WW + p) * D_MODEL + col);
      atomicAdd(&out[gidx], c[i]);
    }
  }
}

extern "C" void kernel_launch(void* const* d_in, const int* in_sizes, int n_in,
                              void* d_out, int out_size, void* d_ws, size_t ws_size,
                              hipStream_t stream) {
  const float* x      = (const float*)d_in[0];
  const float* W_in   = (const float*)d_in[1];
  const float* conv_w = (const float*)d_in[2];
  const float* conv_b = (const float*)d_in[3];
  const float* W_x    = (const float*)d_in[4];
  const float* W_dt   = (const float*)d_in[5];
  const float* b_dt   = (const float*)d_in[6];
  const float* A_log  = (const float*)d_in[7];
  const float* Dv     = (const float*)d_in[8];
  const float* W_out  = (const float*)d_in[9];
  float* out = (float*)d_out;

  hipMemsetAsync(out, 0, (size_t)out_size * sizeof(float), stream);
  hipLaunchKernelGGL(spatial_mamba_dir, dim3(4 * NSEQ), dim3(BLOCK), 0, stream,
                     x, W_in, conv_w, conv_b, W_x, W_dt, b_dt, A_log, Dv, W_out, out);
}